// STECardSelector_46196668235907
// MI455X (gfx1250) — compile-verified
//
#include <hip/hip_runtime.h>
#include <stdint.h>

#define ENC    512
#define HID    512
#define G4     2048
#define NSTEPS 5

typedef float v2f __attribute__((ext_vector_type(2)));
typedef float v8f __attribute__((ext_vector_type(8)));

union V8U { v8f v; float f[8]; };

__device__ __forceinline__ float v8_get(v8f v, int j) { V8U u; u.v = v; return u.f[j]; }
__device__ __forceinline__ float NEGINF() { return -__builtin_inff(); }
__device__ __forceinline__ float sigmoidf(float x) { return 1.0f / (1.0f + __expf(-x)); }

// ---------------- threefry2x32 (20 rounds), matching JAX's PRNG core ----------------
__device__ __forceinline__ void tf2x32(uint32_t k0, uint32_t k1,
                                       uint32_t c0, uint32_t c1,
                                       uint32_t& o0, uint32_t& o1) {
  uint32_t ks2 = k0 ^ k1 ^ 0x1BD11BDAu;
  uint32_t x0 = c0 + k0, x1 = c1 + k1;
#define TFR(r) { x0 += x1; x1 = (x1 << (r)) | (x1 >> (32 - (r))); x1 ^= x0; }
  TFR(13) TFR(15) TFR(26) TFR(6)   x0 += k1;  x1 += ks2 + 1u;
  TFR(17) TFR(29) TFR(16) TFR(24)  x0 += ks2; x1 += k0 + 2u;
  TFR(13) TFR(15) TFR(26) TFR(6)   x0 += k0;  x1 += k1 + 3u;
  TFR(17) TFR(29) TFR(16) TFR(24)  x0 += k1;  x1 += ks2 + 4u;
  TFR(13) TFR(15) TFR(26) TFR(6)   x0 += ks2; x1 += k0 + 5u;
#undef TFR
  o0 = x0; o1 = x1;
}

// split(key(42), 5) -> step key s  (flat counter-mode layout of jax.random.split)
__device__ __forceinline__ void step_key(int step, uint32_t& k0, uint32_t& k1) {
  uint32_t f[2];
  for (int t = 0; t < 2; ++t) {
    int j = 2 * step + t;
    uint32_t o0, o1;
    if (j < NSTEPS) { tf2x32(0u, 42u, (uint32_t)j, (uint32_t)(j + NSTEPS), o0, o1); f[t] = o0; }
    else            { tf2x32(0u, 42u, (uint32_t)(j - NSTEPS), (uint32_t)j, o0, o1); f[t] = o1; }
  }
  k0 = f[0]; k1 = f[1];
}

// gumbel(-log(-log(u))) for logical index j of a `total`-long draw
__device__ __forceinline__ float gumbel_at(uint32_t k0, uint32_t k1, uint32_t j, uint32_t total) {
  uint32_t half = (total + 1u) >> 1;
  uint32_t o0, o1, bits;
  if (j < half) { tf2x32(k0, k1, j, j + half, o0, o1); bits = o0; }
  else          { tf2x32(k0, k1, j - half, j, o0, o1); bits = o1; }
  float u = __uint_as_float((bits >> 9) | 0x3f800000u) - 1.0f;   // [0,1)
  return -__logf(-__logf(u));
}

// ---------------- streaming logsumexp / argmax merges ----------------
__device__ __forceinline__ void lse_merge(float& m, float& s, float om, float os) {
  float nm = fmaxf(m, om);
  float a = (s  > 0.f) ? s  * __expf(m  - nm) : 0.f;
  float b = (os > 0.f) ? os * __expf(om - nm) : 0.f;
  m = nm; s = a + b;
}
__device__ __forceinline__ void amax_merge(float& pm, int& pi, float& pl,
                                           float opm, int opi, float opl) {
  if (opm > pm || (opm == pm && opi != -1 && (pi == -1 || opi < pi))) {
    pm = opm; pi = opi; pl = opl;
  }
}

// ---------------- WMMA fp32 16-row x vector helper ----------------
// acc[j] (lanes 0-15) = dot(mat[r0+j], vec); (lanes 16-31) = dot(mat[r0+8+j], vec)
// A: 16x4 f32 tile (lane -> M=lane&15, Koff=2*(lane>>4)); B: vec replicated over 16 cols.
// NT=true streams rows with non-temporal loads (evict-first; keeps the RT-resident
// prefix of the embedding table live in the 192MB L2 across all 5 passes).
template <bool NT>
__device__ __forceinline__ v8f wmma_rows16(const float* __restrict__ mat, int ld,
                                           int r0, int rmax,
                                           const float* __restrict__ vec, int K, v8f acc) {
  const int lane = threadIdx.x & 31;
  const int m    = lane & 15;
  const int koff = (lane >> 4) << 1;               // 0 or 2
  int row = r0 + m; if (row > rmax) row = rmax;    // clamp keeps EXEC all-ones
  const float* rp = mat + (long)row * ld + koff;
  if (!NT) __builtin_prefetch(rp, 0, 1);
#pragma unroll 8
  for (int k = 0; k < K; k += 4) {
    v2f a, b;
    if (NT) a = __builtin_nontemporal_load((const v2f*)(rp + k));
    else    a = *(const v2f*)(rp + k);
    b[0] = vec[k + koff]; b[1] = vec[k + koff + 1];
    acc = __builtin_amdgcn_wmma_f32_16x16x4_f32(false, a, false, b,
                                                (short)0, acc, false, false);
  }
  return acc;
}
// lanes owning a result row: (lane&15)<8, card-in-tile c = 8*(lane>>4) + (lane&7), val = acc[lane&7]
__device__ __forceinline__ bool lane_owns(int lane, int& c) {
  if ((lane & 15) < 8) { c = ((lane >> 4) << 3) + (lane & 7); return true; }
  return false;
}

// ---------------- workspace layout (float indices) ----------------
#define WS_H0    0
#define WS_C0    512
#define WS_H1    1024
#define WS_C1    1536
#define WS_PROJ  2048
#define WS_XSEL  2560
#define WS_GATES 3072
#define WS_TLP   5120
#define WS_INTS  5124   /* int view: sel[0..4], active at +5 */
#define WS_PART  5136

// ---------------- kernels ----------------
__global__ __launch_bounds__(256) void init_state(float* wsf, int* wsi) {
  for (int i = threadIdx.x; i <= WS_TLP; i += 256) wsf[i] = 0.0f;
  if (threadIdx.x == 0) {
    for (int q = 0; q < NSTEPS; ++q) wsi[WS_INTS + q] = -1;
    wsi[WS_INTS + 5] = 1;   // active
  }
}

// out[r] = dot(mat[r], vec) + bias[r]
__global__ __launch_bounds__(256) void matvec_wmma(const float* __restrict__ mat,
                                                   const float* __restrict__ vec,
                                                   const float* __restrict__ bias,
                                                   float* __restrict__ out,
                                                   int rows, int K) {
  __shared__ float sv[ENC];
  for (int i = threadIdx.x; i < K; i += 256) sv[i] = vec[i];
  __syncthreads();
  int wave = threadIdx.x >> 5, lane = threadIdx.x & 31;
  int r0 = blockIdx.x * 128 + wave * 16;
  v8f acc = {};
  acc = wmma_rows16<false>(mat, K, r0, rows - 1, sv, K, acc);
  int c;
  if (lane_owns(lane, c)) {
    int r = r0 + c;
    if (r < rows) out[r] = v8_get(acc, lane & 7) + bias[r];
  }
}

// gates[r] = dot(wih[r], x) + dot(whh[r], h) + bih[r] + bhh[r]
__global__ __launch_bounds__(256) void lstm_gates_wmma(const float* __restrict__ wih,
                                                       const float* __restrict__ whh,
                                                       const float* __restrict__ bih,
                                                       const float* __restrict__ bhh,
                                                       const float* __restrict__ x,
                                                       const float* __restrict__ h,
                                                       float* __restrict__ gates,
                                                       int rows, int K) {
  __shared__ float sx[ENC];
  __shared__ float sh[HID];
  for (int i = threadIdx.x; i < K; i += 256) { sx[i] = x[i]; sh[i] = h[i]; }
  __syncthreads();
  int wave = threadIdx.x >> 5, lane = threadIdx.x & 31;
  int r0 = blockIdx.x * 128 + wave * 16;
  v8f acc = {};
  acc = wmma_rows16<false>(wih, K, r0, rows - 1, sx, K, acc);
  acc = wmma_rows16<false>(whh, K, r0, rows - 1, sh, K, acc);
  int c;
  if (lane_owns(lane, c)) {
    int r = r0 + c;
    if (r < rows) gates[r] = v8_get(acc, lane & 7) + bih[r] + bhh[r];
  }
}

__global__ __launch_bounds__(256) void lstm_update(const float* __restrict__ gates,
                                                   float* __restrict__ h,
                                                   float* __restrict__ c, int Hd) {
  int j = blockIdx.x * 256 + threadIdx.x;
  if (j < Hd) {
    float gi = gates[j], gf = gates[Hd + j], gg = gates[2 * Hd + j], go = gates[3 * Hd + j];
    float cn = sigmoidf(gf) * c[j] + sigmoidf(gi) * tanhf(gg);
    c[j] = cn;
    h[j] = sigmoidf(go) * tanhf(cn);
  }
}

// one streaming pass: WMMA dot for 16 cards/wave, Gumbel perturb, online lse + argmax.
// Rows < ntRows use RT loads (L2-resident across steps); rows >= ntRows use NT loads.
__global__ __launch_bounds__(256) void score_cards(const float* __restrict__ emb,
                                                   const float* __restrict__ proj,
                                                   const int* __restrict__ sel,
                                                   float* __restrict__ partials,
                                                   int N, int K, int step, int ntRows) {
  __shared__ float sp[ENC];
  __shared__ float wm[8], wsum[8], wpm[8], wpl[8];
  __shared__ int wpi[8];
  for (int i = threadIdx.x; i < K; i += 256) sp[i] = proj[i];
  __syncthreads();

  int wave = threadIdx.x >> 5, lane = threadIdx.x & 31;
  int base = blockIdx.x * 128 + wave * 16;   // wave-uniform tile base
  v8f acc = {};
  if (base >= ntRows)  // ntRows is a multiple of 128 -> branch is wave-uniform
    acc = wmma_rows16<true >(emb, K, base, N - 1, sp, K, acc);
  else
    acc = wmma_rows16<false>(emb, K, base, N - 1, sp, K, acc);

  uint32_t k0, k1; step_key(step, k0, k1);
  float m = NEGINF(), s = 0.f, pm = NEGINF(), pl = 0.f;
  int pi = -1, c;
  if (lane_owns(lane, c)) {
    int card = base + c;
    if (card < N) {
      bool avail = true;
      for (int q = 0; q < NSTEPS; ++q) avail = avail && (sel[q] != card);
      if (avail) {
        float logit = v8_get(acc, lane & 7);
        m = logit; s = 1.0f;
        float g = gumbel_at(k0, k1, (uint32_t)card, (uint32_t)(N + 1));
        pm = logit + g; pi = card; pl = logit;
      }
    }
  }
  for (int off = 16; off > 0; off >>= 1) {
    float om  = __shfl_xor(m,  off, 32);
    float os  = __shfl_xor(s,  off, 32);
    float opm = __shfl_xor(pm, off, 32);
    float opl = __shfl_xor(pl, off, 32);
    int   opi = __shfl_xor(pi, off, 32);
    lse_merge(m, s, om, os);
    amax_merge(pm, pi, pl, opm, opi, opl);
  }
  if (lane == 0) { wm[wave] = m; wsum[wave] = s; wpm[wave] = pm; wpl[wave] = pl; wpi[wave] = pi; }
  __syncthreads();
  if (threadIdx.x == 0) {
    float M = wm[0], S = wsum[0], PM = wpm[0], PL = wpl[0]; int PI = wpi[0];
    for (int w = 1; w < 8; ++w) {
      lse_merge(M, S, wm[w], wsum[w]);
      amax_merge(PM, PI, PL, wpm[w], wpi[w], wpl[w]);
    }
    float* p = partials + (long)blockIdx.x * 5;
    p[0] = M; p[1] = S; p[2] = PM; p[3] = __int_as_float(PI); p[4] = PL;
  }
}

__global__ __launch_bounds__(256) void select_finalize(const float* __restrict__ partials, int nparts,
                                                       const float* __restrict__ proj,
                                                       const float* __restrict__ eos_emb,
                                                       const float* __restrict__ card_emb,
                                                       float* __restrict__ x_sel,
                                                       int* __restrict__ sel,   // [0..4]=picked, [5]=active
                                                       float* __restrict__ tlp,
                                                       int* __restrict__ out_idx,
                                                       float* __restrict__ out_lp,
                                                       int N, int K, int step) {
  __shared__ float r_m[256], r_s[256], r_pm[256], r_pl[256];
  __shared__ int r_pi[256];
  __shared__ float s_eos;
  __shared__ int s_idx;

  // eos similarity
  float part = 0.f;
  for (int i = threadIdx.x; i < K; i += 256) part += proj[i] * eos_emb[i];
  r_m[threadIdx.x] = part;
  __syncthreads();
  for (int st = 128; st > 0; st >>= 1) {
    if (threadIdx.x < st) r_m[threadIdx.x] += r_m[threadIdx.x + st];
    __syncthreads();
  }
  if (threadIdx.x == 0) s_eos = r_m[0];
  __syncthreads();

  // combine per-block partials
  float m = NEGINF(), s = 0.f, pm = NEGINF(), pl = 0.f; int pi = -1;
  for (int t = threadIdx.x; t < nparts; t += 256) {
    const float* p = partials + (long)t * 5;
    lse_merge(m, s, p[0], p[1]);
    amax_merge(pm, pi, pl, p[2], __float_as_int(p[3]), p[4]);
  }
  r_m[threadIdx.x] = m; r_s[threadIdx.x] = s;
  r_pm[threadIdx.x] = pm; r_pl[threadIdx.x] = pl; r_pi[threadIdx.x] = pi;
  __syncthreads();
  for (int st = 128; st > 0; st >>= 1) {
    if (threadIdx.x < st) {
      float M = r_m[threadIdx.x], S = r_s[threadIdx.x];
      float PM = r_pm[threadIdx.x], PL = r_pl[threadIdx.x];
      int PI = r_pi[threadIdx.x];
      lse_merge(M, S, r_m[threadIdx.x + st], r_s[threadIdx.x + st]);
      amax_merge(PM, PI, PL, r_pm[threadIdx.x + st], r_pi[threadIdx.x + st], r_pl[threadIdx.x + st]);
      r_m[threadIdx.x] = M; r_s[threadIdx.x] = S;
      r_pm[threadIdx.x] = PM; r_pl[threadIdx.x] = PL; r_pi[threadIdx.x] = PI;
    }
    __syncthreads();
  }

  if (threadIdx.x == 0) {
    float M = r_m[0], S = r_s[0], PM = r_pm[0], PL = r_pl[0]; int PI = r_pi[0];
    uint32_t k0, k1; step_key(step, k0, k1);
    float el = s_eos;
    lse_merge(M, S, el, 1.0f);
    float eg = el + gumbel_at(k0, k1, (uint32_t)N, (uint32_t)(N + 1));
    amax_merge(PM, PI, PL, eg, N, el);

    float lse = M + __logf(S);
    float lp  = PL - lse;
    int active = sel[5];
    float tot = tlp[0] + (active ? lp : 0.0f);
    int is_eos = (PI == N);
    int take = active && !is_eos;
    out_idx[step] = take ? PI : -1;
    sel[step]     = take ? PI : -1;
    sel[5]        = active && !is_eos;
    tlp[0] = tot;
    *out_lp = tot;
    s_idx = PI;
  }
  __syncthreads();
  int idx = s_idx;
  const float* src = (idx < N) ? (card_emb + (long)idx * K) : eos_emb;
  for (int i = threadIdx.x; i < K; i += 256) x_sel[i] = src[i];
}

// ---------------- host launch ----------------
extern "C" void kernel_launch(void* const* d_in, const int* in_sizes, int n_in,
                              void* d_out, int out_size, void* d_ws, size_t ws_size,
                              hipStream_t stream) {
  const float* ctx   = (const float*)d_in[0];
  const float* cards = (const float*)d_in[1];
  const float* eos   = (const float*)d_in[2];
  const float* ctx_w = (const float*)d_in[3];
  const float* ctx_b = (const float*)d_in[4];
  const float* out_w = (const float*)d_in[5];
  const float* out_b = (const float*)d_in[6];
  const float* wih0  = (const float*)d_in[7];
  const float* whh0  = (const float*)d_in[8];
  const float* bih0  = (const float*)d_in[9];
  const float* bhh0  = (const float*)d_in[10];
  const float* wih1  = (const float*)d_in[11];
  const float* whh1  = (const float*)d_in[12];
  const float* bih1  = (const float*)d_in[13];
  const float* bhh1  = (const float*)d_in[14];

  int N = in_sizes[1] / ENC;

  float* wsf  = (float*)d_ws;
  int*   wsi  = (int*)d_ws;
  float* h0   = wsf + WS_H0;
  float* c0   = wsf + WS_C0;
  float* h1   = wsf + WS_H1;
  float* c1   = wsf + WS_C1;
  float* proj = wsf + WS_PROJ;
  float* xsel = wsf + WS_XSEL;
  float* gts  = wsf + WS_GATES;
  float* tlp  = wsf + WS_TLP;
  int*   sel  = wsi + WS_INTS;
  float* part = wsf + WS_PART;
  int*   oidx = (int*)d_out;
  float* olp  = ((float*)d_out) + NSTEPS;

  int NB = (N + 127) / 128;
  // L2 partition: keep ~80% of the embedding table RT-resident (164MB of 192MB L2,
  // leaving room for the 17MB of LSTM weights re-read every step); stream the rest NT.
  int ntRows = (int)(((long)N * 4 / 5) & ~127L);

  init_state<<<1, 256, 0, stream>>>(wsf, wsi);
  // x0 = ctx_w @ context + ctx_b; then two LSTM cells with zero state (ws zeroed)
  matvec_wmma<<<4, 256, 0, stream>>>(ctx_w, ctx, ctx_b, xsel, ENC, ENC);
  lstm_gates_wmma<<<16, 256, 0, stream>>>(wih0, whh0, bih0, bhh0, xsel, h0, gts, G4, ENC);
  lstm_update<<<2, 256, 0, stream>>>(gts, h0, c0, HID);
  lstm_gates_wmma<<<16, 256, 0, stream>>>(wih1, whh1, bih1, bhh1, h0, h1, gts, G4, HID);
  lstm_update<<<2, 256, 0, stream>>>(gts, h1, c1, HID);

  for (int s = 0; s < NSTEPS; ++s) {
    matvec_wmma<<<4, 256, 0, stream>>>(out_w, h1, out_b, proj, ENC, HID);
    score_cards<<<NB, 256, 0, stream>>>(cards, proj, sel, part, N, ENC, s, ntRows);
    select_finalize<<<1, 256, 0, stream>>>(part, NB, proj, eos, cards, xsel,
                                           sel, tlp, oidx, olp, N, ENC, s);
    lstm_gates_wmma<<<16, 256, 0, stream>>>(wih0, whh0, bih0, bhh0, xsel, h0, gts, G4, ENC);
    lstm_update<<<2, 256, 0, stream>>>(gts, h0, c0, HID);
    lstm_gates_wmma<<<16, 256, 0, stream>>>(wih1, whh1, bih1, bhh1, h0, h1, gts, G4, HID);
    lstm_update<<<2, 256, 0, stream>>>(gts, h1, c1, HID);
  }
  (void)n_in; (void)out_size; (void)ws_size;
}